// Gamba_45535243272926
// MI455X (gfx1250) — compile-verified
//
#include <hip/hip_runtime.h>
#include <math.h>

// ---------------- problem constants (match reference) ----------------
#define HH   128
#define NN   65536
#define EE   524288
#define GG   256
#define TT   4
#define DD   256      // 2*H
#define DI_  512      // 2*D
#define NS_  128
#define DTR_ 16
#define NLAY 8

typedef float v2f __attribute__((ext_vector_type(2)));
typedef float v8f __attribute__((ext_vector_type(8)));

// =====================================================================
// Generic fp32 WMMA GEMM:  C[M,N] = A[M,K] @ B[K,N] (+bias) (+=C) (relu)
//   block = 256 threads (8 waves), tile = 128x64, K staged in 32-chunks.
//   Each wave computes a 16x64 strip with V_WMMA_F32_16X16X4_F32.
//   A tile stride padded even -> a-frag is one aligned ds_load_b64.
//   B tile stored k-pair-interleaved (float2 per (k/2, n)) -> each b-frag
//   is one aligned ds_load_b64, no register repacking.
//   Requires M % 128 == 0 (true for all call sites); N,K arbitrary,
//   zero-padded in LDS.
// =====================================================================
__global__ __launch_bounds__(256) void k_gemm(
    const float* __restrict__ A, int lda,
    const float* __restrict__ B, int ldb,
    float* __restrict__ C, int ldc,
    const float* __restrict__ bias,
    int M, int K, int N, int flags)   // flags: 1=accumulate into C, 2=relu
{
    __shared__ float As[128][34];          // even stride -> 8B-aligned pairs
    __shared__ float Bs2[16 * 130];        // [k/2][n] float2 interleaved (+pad)

    const int t    = threadIdx.x;
    const int wave = t >> 5;
    const int lane = t & 31;
    const int m0   = blockIdx.y * 128;
    const int n0   = blockIdx.x * 64;
    const int rowA = wave * 16 + (lane & 15);
    const int koff = (lane >> 4) * 2;      // 0 for lanes 0-15, 2 for 16-31
    const int ncol = lane & 15;

    v8f c0 = {}; v8f c1 = {}; v8f c2 = {}; v8f c3 = {};

    for (int kc = 0; kc < K; kc += 32) {
        // stage A tile 128x32 (16 elems / thread)
        #pragma unroll
        for (int i = 0; i < 16; ++i) {
            int idx = i * 256 + t;
            int r = idx >> 5, cc = idx & 31;
            int kk = kc + cc;
            As[r][cc] = (kk < K) ? A[(size_t)(m0 + r) * lda + kk] : 0.f;
        }
        // stage B tile 32x64, k-pair interleaved (8 elems / thread)
        #pragma unroll
        for (int i = 0; i < 8; ++i) {
            int idx = i * 256 + t;
            int r = idx >> 6, cc = idx & 63;
            int kk = kc + r, nn = n0 + cc;
            Bs2[(r >> 1) * 130 + cc * 2 + (r & 1)] =
                (kk < K && nn < N) ? B[(size_t)kk * ldb + nn] : 0.f;
        }
        if (kc + 32 < K)
            __builtin_prefetch(&A[(size_t)(m0 + (t >> 1)) * lda + kc + 32], 0, 1);
        __syncthreads();

        #pragma unroll
        for (int kk = 0; kk < 32; kk += 4) {
            v2f a = *(const v2f*)&As[rowA][kk + koff];
            const float* bp = &Bs2[((kk + koff) >> 1) * 130 + ncol * 2];
            v2f b0 = *(const v2f*)(bp);
            v2f b1 = *(const v2f*)(bp + 32);
            v2f b2 = *(const v2f*)(bp + 64);
            v2f b3 = *(const v2f*)(bp + 96);
            c0 = __builtin_amdgcn_wmma_f32_16x16x4_f32(false, a, false, b0, (short)0, c0, false, false);
            c1 = __builtin_amdgcn_wmma_f32_16x16x4_f32(false, a, false, b1, (short)0, c1, false, false);
            c2 = __builtin_amdgcn_wmma_f32_16x16x4_f32(false, a, false, b2, (short)0, c2, false, false);
            c3 = __builtin_amdgcn_wmma_f32_16x16x4_f32(false, a, false, b3, (short)0, c3, false, false);
        }
        __syncthreads();
    }

    // epilogue: (m,n) of D-frag: vgpr r, lane l -> m = r + 8*(l>>4), n = l&15
    const int mbase = m0 + wave * 16 + (lane >> 4) * 8;
    v8f accs[4] = {c0, c1, c2, c3};
    #pragma unroll
    for (int nt = 0; nt < 4; ++nt) {
        int n = n0 + nt * 16 + ncol;
        if (n >= N) continue;
        float bv = bias ? bias[n] : 0.f;
        #pragma unroll
        for (int r = 0; r < 8; ++r) {
            size_t off = (size_t)(mbase + r) * ldc + n;
            float v = accs[nt][r] + bv;
            if (flags & 1) v += C[off];
            if (flags & 2) v = v > 0.f ? v : 0.f;
            C[off] = v;
        }
    }
}

// ------------------- LayerNorm (block per row, W<=256) -------------------
__global__ void k_ln(const float* __restrict__ in, int in_stride,
                     float* __restrict__ out, int out_stride,
                     const float* __restrict__ g, const float* __restrict__ b,
                     int W, int relu)
{
    extern __shared__ float red[];
    const int row = blockIdx.x, c = threadIdx.x;
    const float* rin = in + (size_t)row * in_stride;
    float v = rin[c];
    red[c] = v; __syncthreads();
    for (int s = W >> 1; s > 0; s >>= 1) { if (c < s) red[c] += red[c + s]; __syncthreads(); }
    float mean = red[0] / W; __syncthreads();
    float d = v - mean;
    red[c] = d * d; __syncthreads();
    for (int s = W >> 1; s > 0; s >>= 1) { if (c < s) red[c] += red[c + s]; __syncthreads(); }
    float var = red[0] / W;
    float y = d * rsqrtf(var + 1e-5f) * g[c] + b[c];
    if (relu) y = fmaxf(y, 0.f);
    out[(size_t)row * out_stride + c] = y;
}

__global__ void k_zero(float* __restrict__ p, size_t n) {
    size_t i = (size_t)blockIdx.x * blockDim.x + threadIdx.x;
    if (i < n) p[i] = 0.f;
}

// edge scatter-add: out[dst[e], :] += in[src[e], :]   (width 128, float4)
__global__ void k_scatter(const float* __restrict__ in, float* __restrict__ out,
                          const int* __restrict__ src, const int* __restrict__ dst)
{
    size_t i = (size_t)blockIdx.x * blockDim.x + threadIdx.x;   // E*32
    if (i >= (size_t)EE * 32) return;
    int e  = (int)(i >> 5);
    int c4 = (int)(i & 31) << 2;
    int s = src[e], d = dst[e];
    const float4 v = *(const float4*)(in + (size_t)s * HH + c4);
    float* o = out + (size_t)d * HH + c4;
    atomicAdd(o + 0, v.x); atomicAdd(o + 1, v.y);
    atomicAdd(o + 2, v.z); atomicAdd(o + 3, v.w);
}

// GRU pointwise: pe = (1-z)*tanh(gn + r*hn) + z*pe ; r,z from s[N,256]
__global__ void k_gru(float* __restrict__ pe, const float* __restrict__ s,
                      const float* __restrict__ gn, const float* __restrict__ hn)
{
    size_t i = (size_t)blockIdx.x * blockDim.x + threadIdx.x;
    if (i >= (size_t)NN * HH) return;
    size_t row = i >> 7; int c = (int)(i & 127);
    float rv = 1.f / (1.f + __expf(-s[row * 256 + c]));
    float zv = 1.f / (1.f + __expf(-s[row * 256 + 128 + c]));
    float nv = tanhf(gn[i] + rv * hn[i]);
    pe[i] = (1.f - zv) * nv + zv * pe[i];
}

// alpha[g,t,l] = sum_d concat(h,pe)[g*256+l, d] * theta[d, t]
__global__ void k_alpha(const float* __restrict__ h, const float* __restrict__ pe,
                        const float* __restrict__ theta, float* __restrict__ alpha)
{
    int g = blockIdx.x, l = threadIdx.x;
    size_t node = (size_t)g * 256 + l;
    const float* hr = h + node * HH;
    const float* pr = pe + node * HH;
    float a0 = 0, a1 = 0, a2 = 0, a3 = 0;
    for (int d = 0; d < 128; ++d) {
        float v = hr[d];
        a0 += v * theta[d * 4 + 0]; a1 += v * theta[d * 4 + 1];
        a2 += v * theta[d * 4 + 2]; a3 += v * theta[d * 4 + 3];
    }
    for (int d0 = 0; d0 < 128; ++d0) {
        float v = pr[d0]; int d = 128 + d0;
        a0 += v * theta[d * 4 + 0]; a1 += v * theta[d * 4 + 1];
        a2 += v * theta[d * 4 + 2]; a3 += v * theta[d * 4 + 3];
    }
    alpha[((size_t)g * 4 + 0) * 256 + l] = a0;
    alpha[((size_t)g * 4 + 1) * 256 + l] = a1;
    alpha[((size_t)g * 4 + 2) * 256 + l] = a2;
    alpha[((size_t)g * 4 + 3) * 256 + l] = a3;
}

// aX[g,t,d] = sum_l alpha[g,t,l] * concat(h,pe)[g*256+l, d]
__global__ void k_aX(const float* __restrict__ alpha, const float* __restrict__ h,
                     const float* __restrict__ pe, float* __restrict__ aX)
{
    int idx = blockIdx.x * blockDim.x + threadIdx.x;       // G*T*D = 262144
    int g = idx >> 10, rem = idx & 1023, tt = rem >> 8, d = rem & 255;
    const float* arow = alpha + ((size_t)g * 4 + tt) * 256;
    const float* base = (d < 128) ? (h + d) : (pe + (d - 128));
    size_t node0 = (size_t)g * 256;
    float acc = 0.f;
    for (int l = 0; l < 256; ++l)
        acc += arow[l] * base[(node0 + l) * HH];
    aX[((size_t)g * 4 + tt) * 256 + d] = acc;
}

// causal depthwise conv (DC=4) + SiLU ; xs = uW[:, :512]
__global__ void k_conv_silu(const float* __restrict__ uW, const float* __restrict__ Wc,
                            const float* __restrict__ bc, float* __restrict__ xc)
{
    int idx = blockIdx.x * blockDim.x + threadIdx.x;       // 1024*512
    if (idx >= 1024 * 512) return;
    int row = idx >> 9, c = idx & 511;
    int b = row >> 2, l = row & 3;
    float acc = bc[c];
    #pragma unroll
    for (int k = 0; k < 4; ++k) {
        int ls = l + k - 3;
        if (ls >= 0) acc += uW[((size_t)(b * 4 + ls)) * 1024 + c] * Wc[c * 4 + k];
    }
    xc[idx] = acc / (1.f + __expf(-acc));                  // silu
}

__global__ void k_softplus(float* __restrict__ p, size_t n) {
    size_t i = (size_t)blockIdx.x * blockDim.x + threadIdx.x;
    if (i >= n) return;
    float x = p[i];
    p[i] = (x > 20.f) ? x : log1pf(__expf(x));
}

// selective-scan: one wave per (b, di); 4 states/lane over NS=128.
// y[b,l,di] = (sum_ns h*C + xc*Dp) * silu(z)
__global__ __launch_bounds__(256) void k_scan(
    const float* __restrict__ dt, const float* __restrict__ dbc,
    const float* __restrict__ xc, const float* __restrict__ uW,
    const float* __restrict__ Alog, const float* __restrict__ Dp,
    float* __restrict__ y)
{
    int w = blockIdx.x * 8 + (threadIdx.x >> 5);           // 0 .. 256*512-1
    int lane = threadIdx.x & 31;
    int b = w >> 9, di = w & 511;
    float Aneg[4], hst[4] = {0.f, 0.f, 0.f, 0.f};
    #pragma unroll
    for (int j = 0; j < 4; ++j)
        Aneg[j] = -__expf(Alog[(size_t)di * NS_ + lane + 32 * j]);
    #pragma unroll
    for (int l = 0; l < 4; ++l) {
        size_t row = (size_t)b * 4 + l;
        float dtv = dt[row * 512 + di];
        float xcv = xc[row * 512 + di];
        float sum = 0.f;
        #pragma unroll
        for (int j = 0; j < 4; ++j) {
            int ns = lane + 32 * j;
            float Bv = dbc[row * 272 + 16 + ns];
            float Cv = dbc[row * 272 + 144 + ns];
            float dA = __expf(dtv * Aneg[j]);
            hst[j] = dA * hst[j] + dtv * Bv * xcv;
            sum += hst[j] * Cv;
        }
        #pragma unroll
        for (int off = 16; off > 0; off >>= 1)
            sum += __shfl_xor(sum, off, 32);
        if (lane == 0) {
            float zv  = uW[row * 1024 + 512 + di];
            float sil = zv / (1.f + __expf(-zv));
            y[row * 512 + di] = (sum + xcv * Dp[di]) * sil;
        }
    }
}

// xg[i,:] += tbl[batch[i], :]   (width 128)
__global__ void k_add_gather(float* __restrict__ xg, const float* __restrict__ tbl,
                             const int* __restrict__ batch)
{
    size_t i = (size_t)blockIdx.x * blockDim.x + threadIdx.x;
    if (i >= (size_t)NN * HH) return;
    size_t row = i >> 7; int c = (int)(i & 127);
    xg[i] += tbl[(size_t)batch[row] * HH + c];
}

__global__ void k_add(float* __restrict__ a, const float* __restrict__ b, size_t n) {
    size_t i = (size_t)blockIdx.x * blockDim.x + threadIdx.x;
    if (i < n) a[i] += b[i];
}

// out[g,c] = sum_{l<256} in[g*256+l, c]   (contiguous batch segments)
__global__ void k_groupsum(const float* __restrict__ in, float* __restrict__ out)
{
    int g = blockIdx.x, c = threadIdx.x;                   // 128 threads
    float acc = 0.f;
    for (int l = 0; l < 256; ++l) acc += in[((size_t)g * 256 + l) * HH + c];
    out[(size_t)g * HH + c] = acc;
}

__global__ void k_transpose(const float* __restrict__ in, float* __restrict__ out,
                            int R, int Cc)
{
    int i = blockIdx.x * blockDim.x + threadIdx.x;
    if (i >= R * Cc) return;
    int r = i / Cc, c = i % Cc;
    out[(size_t)c * R + r] = in[i];
}

// =====================================================================
extern "C" void kernel_launch(void* const* d_in, const int* in_sizes, int n_in,
                              void* d_out, int out_size, void* d_ws, size_t ws_size,
                              hipStream_t stream) {
    (void)in_sizes; (void)n_in; (void)out_size; (void)ws_size;
    const float* x       = (const float*)d_in[0];
    const int*   eidx    = (const int*)  d_in[1];
    const int*   batch   = (const int*)  d_in[2];
    const float* enc_W1  = (const float*)d_in[3];
    const float* enc_b1  = (const float*)d_in[4];
    const float* enc_g1  = (const float*)d_in[5];
    const float* enc_bt1 = (const float*)d_in[6];
    const float* enc_W2  = (const float*)d_in[7];
    const float* enc_b2  = (const float*)d_in[8];
    const float* ggc_W   = (const float*)d_in[9];
    const float* gru_Wih = (const float*)d_in[10];
    const float* gru_Whh = (const float*)d_in[11];
    const float* gru_bih = (const float*)d_in[12];
    const float* gru_bhh = (const float*)d_in[13];
    const float* theta_W = (const float*)d_in[14];
    const float* m_inW   = (const float*)d_in[15];
    const float* m_convW = (const float*)d_in[16];
    const float* m_convb = (const float*)d_in[17];
    const float* m_xW    = (const float*)d_in[18];
    const float* m_dtW   = (const float*)d_in[19];
    const float* m_dtb   = (const float*)d_in[20];
    const float* m_Alog  = (const float*)d_in[21];
    const float* m_D     = (const float*)d_in[22];
    const float* m_outW  = (const float*)d_in[23];
    const float* lnm_g   = (const float*)d_in[24];
    const float* lnm_b   = (const float*)d_in[25];
    const float* mrg_W   = (const float*)d_in[26];
    const float* mrg_b   = (const float*)d_in[27];
    const float* og_W    = (const float*)d_in[28];
    const float* og_b    = (const float*)d_in[29];
    const float* dec_W1  = (const float*)d_in[30];
    const float* dec_b1  = (const float*)d_in[31];
    const float* dec_g1  = (const float*)d_in[32];
    const float* dec_bt1 = (const float*)d_in[33];
    const float* dec_W2  = (const float*)d_in[34];
    const float* dec_b2  = (const float*)d_in[35];
    const int* src = eidx;
    const int* dst = eidx + EE;
    float* out = (float*)d_out;

    // ---------------- workspace arena ----------------
    float* ws = (float*)d_ws;
    size_t o = 0;
    auto alloc = [&](size_t n) { float* p = ws + o; o += n; return p; };
    float* R0   = alloc((size_t)NN * 128);   // h (x_orig)
    float* R1   = alloc((size_t)NN * 128);   // pe
    float* R2   = alloc((size_t)NN * 128);   // hW / gn / xg
    float* R3   = alloc((size_t)NN * 128);   // m agg / xg-agg
    float* R4   = alloc((size_t)NN * 256);   // s = gi+gh (first 2H)
    float* R5   = alloc((size_t)NN * 128);   // hn / xg2
    float* WihT = alloc(128 * 384);
    float* WhhT = alloc(128 * 384);
    float* alB  = alloc((size_t)GG * 4 * 256);
    float* aXB  = alloc((size_t)GG * 4 * 256);
    float* uW   = alloc(1024 * 1024);
    float* xcB  = alloc(1024 * 512);
    float* dbcB = alloc(1024 * 272);
    float* dtB  = alloc(1024 * 512);
    float* yB   = alloc(1024 * 512);
    float* xmB  = alloc(1024 * 256);
    float* xlast= alloc(256 * 256);
    float* tmrg = alloc(256 * 128);
    float* outg = alloc(256 * 128);
    float* dec1 = alloc(256 * 128);

    auto gemm = [&](const float* A, int lda, const float* B, int ldb,
                    float* C, int ldc, const float* bias,
                    int M, int K, int Nc, int flags) {
        dim3 grid((Nc + 63) / 64, M / 128);
        k_gemm<<<grid, 256, 0, stream>>>(A, lda, B, ldb, C, ldc, bias, M, K, Nc, flags);
    };
    const size_t NH = (size_t)NN * HH;
    const int ZB = (int)((NH + 255) / 256);

    // ---------------- encoder: h = ReLU(LN(x@W1+b1))@W2+b2 ----------------
    gemm(x, 128, enc_W1, 128, R2, 128, enc_b1, NN, 128, 128, 0);
    k_ln<<<NN, 128, 128 * 4, stream>>>(R2, 128, R2, 128, enc_g1, enc_bt1, 128, 1);
    gemm(R2, 128, enc_W2, 128, R0, 128, enc_b2, NN, 128, 128, 0);
    hipMemcpyAsync(R1, R0, NH * sizeof(float), hipMemcpyDeviceToDevice, stream);

    // ---------------- pre-transpose GRU weights ----------------
    k_transpose<<<(384 * 128 + 255) / 256, 256, 0, stream>>>(gru_Wih, WihT, 384, 128);
    k_transpose<<<(384 * 128 + 255) / 256, 256, 0, stream>>>(gru_Whh, WhhT, 384, 128);

    // ---------------- 8 gated-graph-conv layers ----------------
    for (int layer = 0; layer < NLAY; ++layer) {
        const float* Wl = ggc_W + (size_t)layer * 128 * 128;
        gemm(R1, 128, Wl, 128, R2, 128, nullptr, NN, 128, 128, 0);        // hW = pe@Wl
        k_zero<<<ZB, 256, 0, stream>>>(R3, NH);
        k_scatter<<<(EE * 32 + 255) / 256, 256, 0, stream>>>(R2, R3, src, dst);
        // s = m@WihT[:, :256] + bih[:256]  ;  s += pe@WhhT[:, :256] + bhh[:256]
        gemm(R3, 128, WihT, 384, R4, 256, gru_bih, NN, 128, 256, 0);
        gemm(R1, 128, WhhT, 384, R4, 256, gru_bhh, NN, 128, 256, 1);
        // gn = m@WihT[:, 256:] + bih[256:] ; hn = pe@WhhT[:, 256:] + bhh[256:]
        gemm(R3, 128, WihT + 256, 384, R2, 128, gru_bih + 256, NN, 128, 128, 0);
        gemm(R1, 128, WhhT + 256, 384, R5, 128, gru_bhh + 256, NN, 128, 128, 0);
        k_gru<<<ZB, 256, 0, stream>>>(R1, R4, R2, R5);
    }

    // ---------------- attention pooling: alpha, aX ----------------
    k_alpha<<<GG, 256, 0, stream>>>(R0, R1, theta_W, alB);
    k_aX<<<(GG * 4 * 256) / 256, 256, 0, stream>>>(alB, R0, R1, aXB);

    // ---------------- Mamba block (B=256, L=4) ----------------
    gemm(aXB, 256, m_inW, 1024, uW, 1024, nullptr, 1024, 256, 1024, 0);
    k_conv_silu<<<(1024 * 512) / 256, 256, 0, stream>>>(uW, m_convW, m_convb, xcB);
    gemm(xcB, 512, m_xW, 272, dbcB, 272, nullptr, 1024, 512, 272, 0);
    gemm(dbcB, 272, m_dtW, 512, dtB, 512, m_dtb, 1024, 16, 512, 0);      // dtr slice via lda
    k_softplus<<<(1024 * 512) / 256, 256, 0, stream>>>(dtB, (size_t)1024 * 512);
    k_scan<<<(GG * 512) / 8, 256, 0, stream>>>(dtB, dbcB, xcB, uW, m_Alog, m_D, yB);
    gemm(yB, 512, m_outW, 256, xmB, 256, nullptr, 1024, 512, 256, 0);
    // x_m = LN(xm[:, -1, :])  (only last timestep is consumed downstream)
    k_ln<<<GG, 256, 256 * 4, stream>>>(xmB + 3 * 256, 1024, xlast, 256, lnm_g, lnm_b, 256, 0);

    // ---------------- merge: xg = h@Wtop + (x_m@Wbot + b)[batch] ----------------
    gemm(xlast, 256, mrg_W + 128 * 128, 128, tmrg, 128, mrg_b, 256, 256, 128, 0);
    gemm(R0, 128, mrg_W, 128, R2, 128, nullptr, NN, 128, 128, 0);
    k_add_gather<<<ZB, 256, 0, stream>>>(R2, tmrg, batch);
    // xg = (xg + agg(xg)) @ og_W + og_b
    k_zero<<<ZB, 256, 0, stream>>>(R3, NH);
    k_scatter<<<(EE * 32 + 255) / 256, 256, 0, stream>>>(R2, R3, src, dst);
    k_add<<<ZB, 256, 0, stream>>>(R3, R2, NH);
    gemm(R3, 128, og_W, 128, R5, 128, og_b, NN, 128, 128, 0);
    // per-graph sum (batch segments are contiguous 256-row blocks)
    k_groupsum<<<GG, 128, 0, stream>>>(R5, outg);

    // ---------------- decoder MLP2 ----------------
    gemm(outg, 128, dec_W1, 128, dec1, 128, dec_b1, GG, 128, 128, 0);
    k_ln<<<GG, 128, 128 * 4, stream>>>(dec1, 128, dec1, 128, dec_g1, dec_bt1, 128, 1);
    gemm(dec1, 128, dec_W2, 64, out, 64, dec_b2, GG, 128, 64, 0);
}